// ConfAccLoss_23502061044340
// MI455X (gfx1250) — compile-verified
//
#include <hip/hip_runtime.h>

#define NBINS 15
#define CDIM 128
#define SLOTS 64        // global histogram slots to spread atomic contention
#define PF_ROWS 16      // prefetch distance in rows (8 KB ahead per wave)

typedef float vf4 __attribute__((ext_vector_type(4)));

// Process one 128-wide row held as float4-per-lane across the wave32.
// Packs (correct<<16 | 1) into a single LDS atomic per row.
__device__ __forceinline__ void process_row(vf4 v, int lane, int lab,
                                            unsigned* s_hist) {
    // per-lane max + first-index argmax over the 4 local elements
    float m = v.x; int mi = 0;
    if (v.y > m) { m = v.y; mi = 1; }
    if (v.z > m) { m = v.z; mi = 2; }
    if (v.w > m) { m = v.w; mi = 3; }
    int col = (lane << 2) + mi;

    // wave32 butterfly reduce: max value, min column index on ties
    #pragma unroll
    for (int off = 16; off > 0; off >>= 1) {
        float om = __shfl_xor(m, off, 32);
        int   oc = __shfl_xor(col, off, 32);
        if (om > m || (om == m && oc < col)) { m = om; col = oc; }
    }

    // sum of exp(x - max); the max lane contributes exp(0)=1, so conf = 1/s
    float s = __expf(v.x - m) + __expf(v.y - m) + __expf(v.z - m) + __expf(v.w - m);
    #pragma unroll
    for (int off = 16; off > 0; off >>= 1) s += __shfl_xor(s, off, 32);

    if (lane == 0) {
        float conf = 1.0f / s;
        // bin = searchsorted(linspace(0,1,16)[:-1], conf, 'right') - 1
        // linspace boundaries are i * (1/15) in f32
        const float step = 1.0f / 15.0f;
        int bin = 0;
        #pragma unroll
        for (int i = 1; i < NBINS; ++i) bin += (conf >= (float)i * step) ? 1 : 0;
        unsigned add = 1u | ((col == lab) ? (1u << 16) : 0u);
        atomicAdd(&s_hist[bin], add);
    }
}

__global__ __launch_bounds__(256) void ece_hist_kernel(
    const float* __restrict__ logits, const long long* __restrict__ labels,
    unsigned* __restrict__ gbins, int n, int rpw) {
    __shared__ unsigned s_hist[NBINS];   // [31:16] correct, [15:0] total
    const int tid = threadIdx.x;
    if (tid < NBINS) s_hist[tid] = 0u;
    __syncthreads();

    const int lane = tid & 31;
    const int wid  = blockIdx.x * (blockDim.x >> 5) + (tid >> 5);
    // wid*rpw is wave-uniform: force scalar so row math lands in SALU and
    // labels[row] lowers to s_load_b64 (constant cache) instead of vmem.
    int row = __builtin_amdgcn_readfirstlane(wid * rpw);
    int end = row + rpw; if (end > n) end = n;
    end = __builtin_amdgcn_readfirstlane(end);

    // 4 rows in flight per wave (4 x b128 NT loads back-to-back = 2KB),
    // plus global_prefetch 16 rows ahead to deepen the pipeline.
    for (; row + 4 <= end; row += 4) {
        const vf4* p0 = (const vf4*)(logits + (size_t)(row    ) * CDIM);
        const vf4* p1 = (const vf4*)(logits + (size_t)(row + 1) * CDIM);
        const vf4* p2 = (const vf4*)(logits + (size_t)(row + 2) * CDIM);
        const vf4* p3 = (const vf4*)(logits + (size_t)(row + 3) * CDIM);
        // speculative prefetch (dropped silently if past the allocation)
        __builtin_prefetch((const char*)(p0 + lane) + (size_t)PF_ROWS * CDIM * 4, 0, 1);
        __builtin_prefetch((const char*)(p1 + lane) + (size_t)PF_ROWS * CDIM * 4, 0, 1);
        __builtin_prefetch((const char*)(p2 + lane) + (size_t)PF_ROWS * CDIM * 4, 0, 1);
        __builtin_prefetch((const char*)(p3 + lane) + (size_t)PF_ROWS * CDIM * 4, 0, 1);
        vf4 v0 = __builtin_nontemporal_load(p0 + lane);
        vf4 v1 = __builtin_nontemporal_load(p1 + lane);
        vf4 v2 = __builtin_nontemporal_load(p2 + lane);
        vf4 v3 = __builtin_nontemporal_load(p3 + lane);
        // wave-uniform index -> scalar (s_load) label fetches
        int l0 = (int)labels[row];
        int l1 = (int)labels[row + 1];
        int l2 = (int)labels[row + 2];
        int l3 = (int)labels[row + 3];
        process_row(v0, lane, l0, s_hist);
        process_row(v1, lane, l1, s_hist);
        process_row(v2, lane, l2, s_hist);
        process_row(v3, lane, l3, s_hist);
    }
    for (; row < end; ++row) {
        const vf4* p = (const vf4*)(logits + (size_t)row * CDIM);
        vf4 v = __builtin_nontemporal_load(p + lane);
        process_row(v, lane, (int)labels[row], s_hist);
    }

    __syncthreads();
    if (tid < NBINS) {
        unsigned h = s_hist[tid];
        if (h) {
            unsigned* base = gbins + (unsigned)(blockIdx.x & (SLOTS - 1)) * (NBINS * 2);
            atomicAdd(base + tid, h >> 16);             // correct
            atomicAdd(base + NBINS + tid, h & 0xFFFFu); // total
        }
    }
}

__global__ void ece_final_kernel(const unsigned* __restrict__ gbins,
                                 float* __restrict__ out) {
    int t = threadIdx.x;
    if (t < NBINS) {
        unsigned c = 0, tt = 0;
        #pragma unroll 4
        for (int s = 0; s < SLOTS; ++s) {
            c  += gbins[s * (NBINS * 2) + t];
            tt += gbins[s * (NBINS * 2) + NBINS + t];
        }
        out[2 * t]     = (float)c;          // correct count (exact: < 2^24)
        out[2 * t + 1] = (float)(tt - c);   // incorrect count
    }
}

extern "C" void kernel_launch(void* const* d_in, const int* in_sizes, int n_in,
                              void* d_out, int out_size, void* d_ws, size_t ws_size,
                              hipStream_t stream) {
    const float*     logits = (const float*)d_in[0];
    const long long* labels = (const long long*)d_in[1];
    float*    out   = (float*)d_out;
    unsigned* gbins = (unsigned*)d_ws;
    const int n = in_sizes[1];   // number of rows == number of labels

    hipMemsetAsync(gbins, 0, SLOTS * NBINS * 2 * sizeof(unsigned), stream);

    const int threads = 256;                       // 8 waves per block
    const int blocks  = 4096;                      // ~32K waves
    const int nwaves  = blocks * (threads / 32);
    const int rpw     = (n + nwaves - 1) / nwaves; // rows per wave, computed on host

    ece_hist_kernel<<<blocks, threads, 0, stream>>>(logits, labels, gbins, n, rpw);
    ece_final_kernel<<<1, 32, 0, stream>>>(gbins, out);
}